// Layer_61658550501714
// MI455X (gfx1250) — compile-verified
//
#include <hip/hip_runtime.h>

#define Bdim 32
#define Tdim 1024
#define Ddim 512
#define Udim 256
#define Edim 128
#define Kmix 10
#define PK   30      // 3*K params per timestep

#define TM   16      // timesteps per workgroup (WMMA M)
#define PHI_PITCH 260

typedef float v2f __attribute__((ext_vector_type(2)));
typedef float v8f __attribute__((ext_vector_type(8)));

__global__ __launch_bounds__(256) void attn_window_kernel(
    const float* __restrict__ h, const float* __restrict__ chars,
    const float* __restrict__ W, const float* __restrict__ bias,
    float* __restrict__ out)
{
    __shared__ float s_params[TM][PK];        // [t][a0..a9, b0..b9, k0..k9]
    __shared__ float s_phi[TM][PHI_PITCH];    // padded 16x256 phi tile

    const int tid = threadIdx.x;
    const int wg  = blockIdx.x;
    const int b   = wg / (Tdim / TM);
    const int t0  = (wg % (Tdim / TM)) * TM;

    // ---------------- Phase A: params = h @ W + bias for 16 rows ----------------
    for (int idx = tid; idx < TM * PK; idx += 256) {
        const int r = idx / PK;
        const int k = idx - r * PK;
        const float4* hrow = (const float4*)(h + (size_t)(b * Tdim + t0 + r) * Ddim);
        float acc = bias[k];
        #pragma unroll 4
        for (int d4 = 0; d4 < Ddim / 4; ++d4) {
            const float4 hv = hrow[d4];
            const int d = d4 * 4;
            acc += hv.x * W[(d + 0) * PK + k];
            acc += hv.y * W[(d + 1) * PK + k];
            acc += hv.z * W[(d + 2) * PK + k];
            acc += hv.w * W[(d + 3) * PK + k];
        }
        s_params[r][k] = acc;
    }
    __syncthreads();

    // ---------------- Phase B: phi[t][u] = sum_k a*exp(-beta*(kappa-u)^2) -------
    for (int idx = tid; idx < TM * Udim; idx += 256) {
        const int t = idx >> 8;          // idx / 256
        const int u = idx & (Udim - 1);
        const float fu = (float)u;
        float acc = 0.0f;
        #pragma unroll
        for (int k = 0; k < Kmix; ++k) {
            const float a  = s_params[t][k];
            const float be = s_params[t][Kmix + k];
            const float ka = s_params[t][2 * Kmix + k];
            const float df = ka - fu;
            acc += a * __expf(-be * df * df);
        }
        s_phi[t][u] = acc;
    }
    __syncthreads();

    // ---------------- Phase C: out tile = phi (16x256) @ chars[b] (256x128) ----
    // 8 waves; wave w computes the 16x16 output tile at columns [16w, 16w+16).
    const int wave = tid >> 5;
    const int lane = tid & 31;
    const int n0   = wave * 16;
    const int half = lane >> 4;          // 0: lanes 0-15, 1: lanes 16-31
    const int l15  = lane & 15;
    const int koff = half * 2;           // A/B K sub-offset per lane half

    const float* __restrict__ charsB = chars + (size_t)b * Udim * Edim;

    v8f c = {};
    for (int ku = 0; ku < Udim; ku += 4) {
        // A fragment (16x4 f32): lane<16 -> M=lane, K=ku+{0,1}; lane>=16 -> K=ku+{2,3}
        v2f a;
        a.x = s_phi[l15][ku + koff];
        a.y = s_phi[l15][ku + koff + 1];
        // B fragment (4x16 f32): VGPR0 holds K rows {0|2}, VGPR1 holds {1|3}, N=l15
        v2f bf;
        bf.x = charsB[(size_t)(ku + koff)     * Edim + n0 + l15];
        bf.y = charsB[(size_t)(ku + koff + 1) * Edim + n0 + l15];
        c = __builtin_amdgcn_wmma_f32_16x16x4_f32(
                /*neg_a=*/false, a, /*neg_b=*/false, bf,
                /*c_mod=*/(short)0, c, /*reuse_a=*/false, /*reuse_b=*/false);
    }

    // C/D layout: VGPR i -> M = half*8 + i, N = l15
    float* orow = out + (size_t)(b * Tdim + t0 + half * 8) * Edim + n0 + l15;
    #pragma unroll
    for (int i = 0; i < 8; ++i) {
        orow[(size_t)i * Edim] = c[i];
    }
}

extern "C" void kernel_launch(void* const* d_in, const int* in_sizes, int n_in,
                              void* d_out, int out_size, void* d_ws, size_t ws_size,
                              hipStream_t stream) {
    const float* h     = (const float*)d_in[0];
    const float* chars = (const float*)d_in[1];
    const float* W     = (const float*)d_in[2];
    const float* bias  = (const float*)d_in[3];
    float* out = (float*)d_out;

    const dim3 grid(Bdim * (Tdim / TM));   // 32 * 64 = 2048 workgroups
    attn_window_kernel<<<grid, 256, 0, stream>>>(h, chars, W, bias, out);
}